// AssociativeLeaky_55576876810895
// MI455X (gfx1250) — compile-verified
//
#include <hip/hip_runtime.h>
#include <hip/hip_bf16.h>

typedef __attribute__((ext_vector_type(2))) float v2f;
typedef __attribute__((ext_vector_type(8))) float v8f;

#define T_DIM   1024
#define B_DIM   8
#define IN_DIMC 512
#define ROWS    (T_DIM * B_DIM)   // 8192
#define EPSF    1e-8f

// ---------------------------------------------------------------------------
// Kernel 1: fused projection GEMM  [8192 x 512] x [512 x 192] via
// V_WMMA_F32_16X16X4_F32.  Each wave computes one 16-row strip across all
// 192 output columns (12 accumulator tiles: 4 for v, 4 for k, 4 for alpha).
// Epilogue adds bias; alpha tiles get sigmoid.  512 strips = 64 blocks x 8
// waves.  EXEC is always all-ones (full blocks, no divergence) as WMMA
// requires.
// ---------------------------------------------------------------------------
__global__ __launch_bounds__(256) void proj_wmma_kernel(
    const float* __restrict__ x,
    const float* __restrict__ Wv, const float* __restrict__ bv,
    const float* __restrict__ Wk, const float* __restrict__ bk,
    const float* __restrict__ Wa, const float* __restrict__ ba,
    float* __restrict__ Vp, float* __restrict__ Kp, float* __restrict__ Ap)
{
    const int wave  = threadIdx.x >> 5;
    const int lane  = threadIdx.x & 31;
    const int strip = blockIdx.x * 8 + wave;      // 0..511
    const int row0  = strip * 16;
    const int half  = lane >> 4;                  // 0: lanes 0-15, 1: lanes 16-31
    const int lm    = lane & 15;                  // m (A) or n (B/C) within tile

    v8f acc[12] = {};                             // 12 x 16x16 f32 accumulators

    // A fragment source: x[row0+lm][k0 + 2*half + {0,1}]
    const float* aptr = x + (size_t)(row0 + lm) * IN_DIMC + 2 * half;

    // B fragment sources: B[k][n] = W[n][k]; weight rows are contiguous (512 f32)
    const float* wptr[12];
#pragma unroll
    for (int j = 0; j < 12; ++j) {
        const float* W = (j < 4) ? Wv : ((j < 8) ? Wk : Wa);
        const int nbase = (j & 3) * 16;
        wptr[j] = W + (size_t)(nbase + lm) * IN_DIMC + 2 * half;
    }

    for (int k0 = 0; k0 < IN_DIMC; k0 += 4) {
        v2f af = *(const v2f*)(aptr + k0);
#pragma unroll
        for (int j = 0; j < 12; ++j) {
            v2f bf = *(const v2f*)(wptr[j] + k0);
            acc[j] = __builtin_amdgcn_wmma_f32_16x16x4_f32(
                /*neg_a=*/false, af, /*neg_b=*/false, bf,
                /*c_mod=*/(short)0, acc[j],
                /*reuse_a=*/false, /*reuse_b=*/false);
        }
    }

    // Epilogue: C/D layout -> element acc[j][r] is (M = r + 8*half, N = lm)
#pragma unroll
    for (int j = 0; j < 12; ++j) {
        const int col = (j & 3) * 16 + lm;        // 0..63 within v/k/alpha
        if (j < 4) {
            const float bias = bv[col];
#pragma unroll
            for (int r = 0; r < 8; ++r) {
                const size_t row = (size_t)(row0 + r + 8 * half);
                Vp[row * 64 + col] = acc[j][r] + bias;
            }
        } else if (j < 8) {
            const float bias = bk[col];
#pragma unroll
            for (int r = 0; r < 8; ++r) {
                const size_t row = (size_t)(row0 + r + 8 * half);
                Kp[row * 64 + col] = acc[j][r] + bias;
            }
        } else {
            const float bias = ba[col];
#pragma unroll
            for (int r = 0; r < 8; ++r) {
                const size_t row = (size_t)(row0 + r + 8 * half);
                const float s = acc[j][r] + bias;
                Ap[row * 64 + col] = 1.0f / (1.0f + expf(-s));  // sigmoid
            }
        }
    }
}

// ---------------------------------------------------------------------------
// Kernel 2: leaky-integrator scan.  32768 independent (b,d,n) sequences, one
// thread each, sequential over T -- identical rounding order to the
// reference's cumprod/cumsum formulation:
//   P_t = P_{t-1} * a_t ; acc_t = acc_{t-1} + v_t*k_t/(P_t+eps) ; S_t = acc_t*P_t
// Stores are 32-lane-contiguous (128B) -- this kernel is pure store-BW bound
// (256 MB of output ~= 11 us at 23.3 TB/s).
// ---------------------------------------------------------------------------
__global__ __launch_bounds__(256) void scan_kernel(
    const float* __restrict__ Vp, const float* __restrict__ Kp,
    const float* __restrict__ Ap,
    float* __restrict__ spk, float* __restrict__ mem)
{
    const int tid = blockIdx.x * 256 + threadIdx.x;   // 0..32767
    const int b   = tid >> 12;                        // 0..7
    const int rem = tid & 4095;                       // d*64 + n
    const int d   = rem >> 6;
    const int n   = rem & 63;

    float P = 1.0f, acc = 0.0f;
    for (int t = 0; t < T_DIM; ++t) {
        const int row = t * B_DIM + b;
        // prefetch next timestep's k/alpha rows (global_prefetch_b8)
        if (t + 1 < T_DIM) {
            __builtin_prefetch(Kp + (size_t)(row + B_DIM) * 64 + n, 0, 0);
            __builtin_prefetch(Ap + (size_t)(row + B_DIM) * 64 + n, 0, 0);
        }
        const float v = Vp[(size_t)row * 64 + d];     // broadcast within wave
        const float k = Kp[(size_t)row * 64 + n];     // coalesced
        const float a = Ap[(size_t)row * 64 + n];     // coalesced

        P *= a;
        const float inv = 1.0f / (P + EPSF);
        acc += v * k * inv;
        const float S = acc * P;

        const size_t oi = (size_t)row * 4096 + rem;   // coalesced 128B stores
        mem[oi] = S;
        spk[oi] = (S > 1.0f) ? 1.0f : 0.0f;
    }
}

// ---------------------------------------------------------------------------
extern "C" void kernel_launch(void* const* d_in, const int* in_sizes, int n_in,
                              void* d_out, int out_size, void* d_ws, size_t ws_size,
                              hipStream_t stream)
{
    const float* x  = (const float*)d_in[0];
    const float* Wv = (const float*)d_in[1];
    const float* bv = (const float*)d_in[2];
    const float* Wk = (const float*)d_in[3];
    const float* bk = (const float*)d_in[4];
    const float* Wa = (const float*)d_in[5];
    const float* ba = (const float*)d_in[6];

    float* out = (float*)d_out;
    float* spk = out;                                  // first output
    float* mem = out + (size_t)ROWS * 4096;            // second output

    float* Vp = (float*)d_ws;                          // 8192 x 64
    float* Kp = Vp + (size_t)ROWS * 64;                // 8192 x 64
    float* Ap = Kp + (size_t)ROWS * 64;                // 8192 x 64

    proj_wmma_kernel<<<64, 256, 0, stream>>>(x, Wv, bv, Wk, bk, Wa, ba, Vp, Kp, Ap);
    scan_kernel<<<128, 256, 0, stream>>>(Vp, Kp, Ap, spk, mem);
}